// Loss_1_8323646620405
// MI455X (gfx1250) — compile-verified
//
#include <hip/hip_runtime.h>

typedef float v2f __attribute__((ext_vector_type(2)));
typedef float v8f __attribute__((ext_vector_type(8)));

#define W0c 0.51f
#define W1c 19.05f

// ---------------------------------------------------------------------------
// Wave32 reduction using V_WMMA_F32_16X16X4_F32.
// A = all ones (16x4), B holds the 32 lane accumulators (4x16, rest zero).
// D[m][n] = sum_k B[k][n] = column sums of B. D layout: lane l holds N=l%16,
// identical for every VGPR m. Summing d[0] over all 32 lanes counts each
// column sum exactly twice -> wave sum = 0.5 * full xor-shuffle reduce.
// Robust to the exact B lane->(K,N) mapping: any placement of each lane's
// value exactly once still yields column sums whose total is the wave total.
// ---------------------------------------------------------------------------
__device__ __forceinline__ float wave_reduce_wmma(float v) {
  v2f a; a[0] = 1.0f; a[1] = 1.0f;   // A = ones
  v2f b; b[0] = v;    b[1] = 0.0f;   // B row(s) = lane accumulators
  v8f c = {};
  c = __builtin_amdgcn_wmma_f32_16x16x4_f32(
      /*neg_a=*/false, a, /*neg_b=*/false, b,
      /*c_mod=*/(short)0, c, /*reuse_a=*/false, /*reuse_b=*/false);
  float col = c[0];
  col += __shfl_xor(col, 1, 32);
  col += __shfl_xor(col, 2, 32);
  col += __shfl_xor(col, 4, 32);
  col += __shfl_xor(col, 8, 32);
  col += __shfl_xor(col, 16, 32);
  return 0.5f * col;
}

// Block of 256 threads = 8 waves. All threads must reach this (EXEC all 1s).
__device__ __forceinline__ float block_reduce_256(float v) {
  __shared__ float sm[8];
  float ws = wave_reduce_wmma(v);
  int wid = threadIdx.x >> 5;
  if ((threadIdx.x & 31) == 0) sm[wid] = ws;
  __syncthreads();
  return sm[0] + sm[1] + sm[2] + sm[3] + sm[4] + sm[5] + sm[6] + sm[7];
}

// Per-element loss. 0/1 labels folded into the log arguments -> 5 logf total.
__device__ __forceinline__ float per_elem(
    int t0, int t1, int t2, int t3, int t4c, int t5, int t6, int t7, int t8,
    float sv, float plv, float hdv,
    float p0, float p1, float p2,
    float q0, float q1, float q2, float q3) {
  int any = t0 | t1 | t2 | t3 | t4c | t5 | t6 | t7 | t8;
  bool ys  = (any != 0);
  float w  = ys ? W1c : W0c;
  float xs = ys ? sv : (1.0f - sv);                 // stroke BCE argument
  float xp = t0 ? (1.0f - plv) : plv;               // player BCE argument
  float xh = t7 ? (1.0f - hdv) : hdv;               // hand BCE argument
  float pg = t4c ? p0 : (t5 ? p1 : p2);             // gathered point prob
  float qg = t2 ? q0 : (t3 ? q1 : (t6 ? q2 : q3));  // gathered serve prob
  float r     = w * logf(xs);
  float inner = logf(xp) + logf(xh) + logf(pg) + logf(qg);
  return -(r + (ys ? inner : 0.0f));
}

// ---------------------------------------------------------------------------
// Kernel 1: grid-stride over quads of 4 consecutive elements so every global
// access is a 16B-aligned b128 load (36 target ints = 9 x int4; 144 = 9*16).
// One f32 partial per block into d_ws.
// ---------------------------------------------------------------------------
__global__ void __launch_bounds__(256)
loss_partial_kernel(const float* __restrict__ ps,
                    const float* __restrict__ pp,
                    const float* __restrict__ ph,
                    const float* __restrict__ ppt,
                    const float* __restrict__ psv,
                    const int*   __restrict__ tg,
                    float* __restrict__ partial,
                    int nq) {
  const float4* ps4  = reinterpret_cast<const float4*>(ps);
  const float4* pp4  = reinterpret_cast<const float4*>(pp);
  const float4* ph4  = reinterpret_cast<const float4*>(ph);
  const float4* ppt4 = reinterpret_cast<const float4*>(ppt);
  const float4* psv4 = reinterpret_cast<const float4*>(psv);
  const int4*   tg4  = reinterpret_cast<const int4*>(tg);

  float acc = 0.0f;
  int tid    = blockIdx.x * blockDim.x + threadIdx.x;
  int stride = gridDim.x * blockDim.x;

  for (int q = tid; q < nq; q += stride) {
    float4 s  = ps4[q];
    float4 pl = pp4[q];
    float4 hd = ph4[q];
    float4 a0 = ppt4[3 * q + 0];
    float4 a1 = ppt4[3 * q + 1];
    float4 a2 = ppt4[3 * q + 2];
    float4 b0 = psv4[4 * q + 0];
    float4 b1 = psv4[4 * q + 1];
    float4 b2 = psv4[4 * q + 2];
    float4 b3 = psv4[4 * q + 3];

    int4 tv[9];
#pragma unroll
    for (int j = 0; j < 9; ++j) tv[j] = tg4[9 * q + j];
    const int* t = reinterpret_cast<const int*>(tv);

    acc += per_elem(t[0], t[1], t[2], t[3], t[4], t[5], t[6], t[7], t[8],
                    s.x, pl.x, hd.x, a0.x, a0.y, a0.z, b0.x, b0.y, b0.z, b0.w);
    acc += per_elem(t[9], t[10], t[11], t[12], t[13], t[14], t[15], t[16], t[17],
                    s.y, pl.y, hd.y, a0.w, a1.x, a1.y, b1.x, b1.y, b1.z, b1.w);
    acc += per_elem(t[18], t[19], t[20], t[21], t[22], t[23], t[24], t[25], t[26],
                    s.z, pl.z, hd.z, a1.z, a1.w, a2.x, b2.x, b2.y, b2.z, b2.w);
    acc += per_elem(t[27], t[28], t[29], t[30], t[31], t[32], t[33], t[34], t[35],
                    s.w, pl.w, hd.w, a2.y, a2.z, a2.w, b3.x, b3.y, b3.z, b3.w);
  }

  float bsum = block_reduce_256(acc);
  if (threadIdx.x == 0) partial[blockIdx.x] = bsum;
}

// ---------------------------------------------------------------------------
// Kernel 2: single block folds the per-block partials deterministically and
// scales by 1/N (exact power of two).
// ---------------------------------------------------------------------------
__global__ void __launch_bounds__(256)
reduce_final_kernel(const float* __restrict__ partial, int nb, float scale,
                    float* __restrict__ out) {
  float acc = 0.0f;
  for (int i = threadIdx.x; i < nb; i += blockDim.x) acc += partial[i];
  float tot = block_reduce_256(acc);
  if (threadIdx.x == 0) out[0] = tot * scale;
}

extern "C" void kernel_launch(void* const* d_in, const int* in_sizes, int n_in,
                              void* d_out, int out_size, void* d_ws, size_t ws_size,
                              hipStream_t stream) {
  const float* ps  = (const float*)d_in[0];  // y_pred_stroke  [B,S,1]
  const float* pp  = (const float*)d_in[1];  // y_pred_player  [B,S,1]
  const float* ph  = (const float*)d_in[2];  // y_pred_hand    [B,S,1]
  const float* ppt = (const float*)d_in[3];  // y_pred_point   [B,S,3]
  const float* psv = (const float*)d_in[4];  // y_pred_serve   [B,S,4]
  const int*   tg  = (const int*)d_in[5];    // y_target       [B,S,9]

  const int n  = in_sizes[0];  // B*S elements (2,097,152)
  const int nq = n >> 2;       // quads of 4 elements

  int nb = 1024;               // partials (4 KB of d_ws)
  size_t cap = ws_size / sizeof(float);
  if ((size_t)nb > cap) nb = (int)cap;
  if (nb < 1) nb = 1;
  float* partial = (float*)d_ws;

  hipLaunchKernelGGL(loss_partial_kernel, dim3(nb), dim3(256), 0, stream,
                     ps, pp, ph, ppt, psv, tg, partial, nq);
  hipLaunchKernelGGL(reduce_final_kernel, dim3(1), dim3(256), 0, stream,
                     partial, nb, 1.0f / (float)n, (float*)d_out);
}